// GraphConvolution_13692355740268
// MI455X (gfx1250) — compile-verified
//
#include <hip/hip_runtime.h>
#include <hip/hip_bf16.h>

typedef __attribute__((ext_vector_type(2))) float        v2f;
typedef __attribute__((ext_vector_type(8))) float        v8f;
typedef __attribute__((ext_vector_type(4))) unsigned int v4u;
typedef __attribute__((ext_vector_type(8))) int          v8i;
typedef __attribute__((ext_vector_type(4))) int          v4i;

#define D_IN  128
#define D_OUT 64

// W packed pair-major: Wp[pr][n] = {W[2*pr][n], W[2*pr+1][n]}, row stride
// padded to 80 float2 (=160 dwords, 160 mod 64 == 32) so the two half-wave
// rows hit disjoint LDS bank halves -> conflict-free ds_load_b64.
#define WP_STRIDE 80
#define WP_ROWS   (D_IN / 2)                 // 64 pair-rows
#define WP_ELEMS  (WP_ROWS * WP_STRIDE)      // 5120 float2 = 40 KB

// ---------------------------------------------------------------------------
// Kernel 0: pack W (row-major 128x64) into padded pair-major Wp (one-shot,
// 40 KB). Pad columns zero-filled for determinism.
// ---------------------------------------------------------------------------
__global__ __launch_bounds__(256) void gcn_pack_w(
    const float* __restrict__ W, float2* __restrict__ Wp)
{
    int i = blockIdx.x * 256 + threadIdx.x;
    if (i >= WP_ELEMS) return;
    int pr = i / WP_STRIDE;
    int n  = i % WP_STRIDE;
    float2 v = {0.0f, 0.0f};
    if (n < D_OUT) {
        v.x = W[(2 * pr)     * D_OUT + n];
        v.y = W[(2 * pr + 1) * D_OUT + n];
    }
    Wp[i] = v;
}

// ---------------------------------------------------------------------------
// Kernel 1: support = x @ W via V_WMMA_F32_16X16X4_F32. Wp is staged into LDS
// once per workgroup with the Tensor Data Mover (tensor_load_to_lds), B
// fragments then come from LDS as ds_load_b64. One wave owns a 16x64 tile.
// ---------------------------------------------------------------------------
__global__ __launch_bounds__(128) void gcn_gemm_wmma(
    const float* __restrict__ x, const float2* __restrict__ Wp,
    float* __restrict__ support, int n_mtiles)
{
    __shared__ v2f ldsW[WP_ELEMS];           // 40 KB

    const int lane = threadIdx.x & 31;
    const int wave = threadIdx.x >> 5;

    // ---- TDM: copy 40 KB Wp (global) -> LDS, issued by wave 0 only --------
    if (wave == 0) {
        unsigned long long ga = (unsigned long long)(uintptr_t)Wp;
        unsigned int lds_off  = (unsigned int)(uintptr_t)(&ldsW[0]);

        v4u g0;
        g0.x = 1u;                                        // count=1 (valid D#)
        g0.y = lds_off;                                   // lds_addr [63:32]
        g0.z = (unsigned int)ga;                          // global_addr lo
        g0.w = (unsigned int)((ga >> 32) & 0x1FFFFFFu)    // global_addr[56:32]
             | (2u << 30);                                // type = 2 ("image")

        v8i g1;
        g1[0] = (int)(3u << 16);                          // data_size = 3 (8B)
        g1[1] = (int)(((unsigned)WP_ELEMS & 0xFFFFu) << 16);   // tdim0 lo16
        g1[2] = (int)((((unsigned)WP_ELEMS >> 16) & 0xFFFFu)   // tdim0 hi16
             | (1u << 16));                               // tensor_dim1 = 1
        g1[3] = (int)(((unsigned)WP_ELEMS & 0xFFFFu) << 16);   // tile_dim0
        g1[4] = 1;                                        // tile_dim1 = 1
        g1[5] = WP_ELEMS;                                 // dim0_stride lo32
        g1[6] = 0;
        g1[7] = 0;

        v4i z4 = {0, 0, 0, 0};                            // <=2D: groups 2/3 null
        v8i z8 = {0, 0, 0, 0, 0, 0, 0, 0};
        __builtin_amdgcn_tensor_load_to_lds(g0, g1, z4, z4, z8, 0);
        __builtin_amdgcn_s_wait_tensorcnt(0);
    }
    __syncthreads();

    const int mtile = blockIdx.x * 4 + wave;              // wave-uniform
    if (mtile >= n_mtiles) return;                        // uniform exit

    const int half  = lane >> 4;                          // K-pair selector
    const int l16   = lane & 15;
    const int mbase = mtile * 16;

    v8f acc0 = {}, acc1 = {}, acc2 = {}, acc3 = {};
    const float* __restrict__ xrow = x + (size_t)(mbase + l16) * D_IN;

    for (int k = 0; k < D_IN; k += 4) {
        const int kk = k + 2 * half;
        v2f a;
        a.x = xrow[kk];
        a.y = xrow[kk + 1];

        const v2f* wp = &ldsW[(kk >> 1) * WP_STRIDE];     // ds_load_b64 x4
        v2f b0 = wp[l16];
        v2f b1 = wp[16 + l16];
        v2f b2 = wp[32 + l16];
        v2f b3 = wp[48 + l16];

        acc0 = __builtin_amdgcn_wmma_f32_16x16x4_f32(false, a, false, b0,
                                                     (short)0, acc0, false, false);
        acc1 = __builtin_amdgcn_wmma_f32_16x16x4_f32(false, a, false, b1,
                                                     (short)0, acc1, false, false);
        acc2 = __builtin_amdgcn_wmma_f32_16x16x4_f32(false, a, false, b2,
                                                     (short)0, acc2, false, false);
        acc3 = __builtin_amdgcn_wmma_f32_16x16x4_f32(false, a, false, b3,
                                                     (short)0, acc3, false, false);
    }

    // C/D layout: VGPR r -> row (r + 8*half), col = l16 within each 16-col tile
    float* __restrict__ outp = support + (size_t)mbase * D_OUT;
#pragma unroll
    for (int r = 0; r < 8; ++r) {
        const int m = r + 8 * half;
        outp[(size_t)m * D_OUT +      l16] = acc0[r];
        outp[(size_t)m * D_OUT + 16 + l16] = acc1[r];
        outp[(size_t)m * D_OUT + 32 + l16] = acc2[r];
        outp[(size_t)m * D_OUT + 48 + l16] = acc3[r];
    }
}

// ---------------------------------------------------------------------------
// Kernel 2: out[n][j] = b[j]   (full init; harness poisons d_out)
// ---------------------------------------------------------------------------
__global__ __launch_bounds__(256) void gcn_init_bias(
    float* __restrict__ out, const float* __restrict__ b, int total)
{
    int i = blockIdx.x * 256 + threadIdx.x;
    if (i < total) out[i] = b[i & (D_OUT - 1)];
}

// ---------------------------------------------------------------------------
// Kernel 3: edge scatter. 64 consecutive lanes handle one edge's 64 columns:
// contiguous 256B gather of support[src] and 256B atomic burst to out[dst].
// ---------------------------------------------------------------------------
__global__ __launch_bounds__(256) void gcn_edge_scatter(
    const float* __restrict__ support,
    const int* __restrict__ esrc, const int* __restrict__ edst,
    const float* __restrict__ evals,
    float* __restrict__ out, int n_edges)
{
    unsigned int gid = blockIdx.x * 256u + threadIdx.x;
    int e = (int)(gid >> 6);
    int j = (int)(gid & 63u);
    if (e >= n_edges) return;

    const int   s = esrc[e];
    const int   d = edst[e];
    const float v = evals[e];
    const float msg = v * support[(size_t)s * D_OUT + j];
    unsafeAtomicAdd(&out[(size_t)d * D_OUT + j], msg);
}

// ---------------------------------------------------------------------------
// Kernel 4: in-place ReLU
// ---------------------------------------------------------------------------
__global__ __launch_bounds__(256) void gcn_relu(float* __restrict__ out, int total)
{
    int i = blockIdx.x * 256 + threadIdx.x;
    if (i < total) out[i] = fmaxf(out[i], 0.0f);
}

// ---------------------------------------------------------------------------
extern "C" void kernel_launch(void* const* d_in, const int* in_sizes, int n_in,
                              void* d_out, int out_size, void* d_ws, size_t ws_size,
                              hipStream_t stream)
{
    const float* x     = (const float*)d_in[0];
    const int*   esrc  = (const int*)  d_in[1];
    const int*   edst  = (const int*)  d_in[2];
    const float* evals = (const float*)d_in[3];
    const float* W     = (const float*)d_in[4];
    const float* b     = (const float*)d_in[5];
    float*       out   = (float*)d_out;

    const int n_nodes = in_sizes[0] / D_IN;
    const int n_edges = in_sizes[1];
    const int total   = n_nodes * D_OUT;

    // workspace layout: [ support : n_nodes*64 f32 ][ Wp : 5120 float2 ]
    float*  support = (float*)d_ws;
    float2* Wp      = (float2*)((char*)d_ws + (size_t)total * sizeof(float));

    // 0) pack W into padded pair-major layout (40 KB)
    gcn_pack_w<<<(WP_ELEMS + 255) / 256, 256, 0, stream>>>(W, Wp);

    // 1) support = x @ W  (TDM-staged B in LDS + fp32 WMMA)
    const int n_mtiles = n_nodes / 16;                // 50000/16 = 3125 exact
    const int gemm_blocks = (n_mtiles + 3) / 4;       // 4 waves/block
    gcn_gemm_wmma<<<gemm_blocks, 128, 0, stream>>>(x, Wp, support, n_mtiles);

    // 2) out = bias
    gcn_init_bias<<<(total + 255) / 256, 256, 0, stream>>>(out, b, total);

    // 3) out[dst] += vals * support[src]   (f32 global atomics)
    const long long work = (long long)n_edges * D_OUT;
    const int scat_blocks = (int)((work + 255) / 256);
    gcn_edge_scatter<<<scat_blocks, 256, 0, stream>>>(support, esrc, edst, evals,
                                                      out, n_edges);

    // 4) out = relu(out)
    gcn_relu<<<(total + 255) / 256, 256, 0, stream>>>(out, total);
}